// deforming_medium_34849364639758
// MI455X (gfx1250) — compile-verified
//
#include <hip/hip_runtime.h>
#include <cmath>

#define WW 512
#define NB 32
#define NC 3

typedef float v2f __attribute__((ext_vector_type(2)));
typedef float v8f __attribute__((ext_vector_type(8)));

constexpr float kSCALE  = (float)(4.0 / 511.0);
constexpr float kDEFOFF = (float)(1.0 + 2.0 / 511.0);   // (2 + SCALE)/2
constexpr float kCLIP   = 0.03125f;                     // PIX_W * EPS = (2/512)*8

__device__ __forceinline__ int reflect_idx(int i) {
    i = (i < 0) ? -i : i;
    i = (i >= WW) ? (2 * WW - 2 - i) : i;
    return i;
}

__device__ __forceinline__ float idenv(int i) {
    return (2.0f * (float)i - 511.0f) * (1.0f / 511.0f);
}

__device__ __forceinline__ float wsel(int d, float w0, float w1, float w2, float w3, float w4) {
    float r = 0.0f;
    r = (d == 0) ? w0 : r;
    r = (d == 1) ? w1 : r;
    r = (d == 2) ? w2 : r;
    r = (d == 3) ? w3 : r;
    r = (d == 4) ? w4 : r;
    return r;
}

__device__ __forceinline__ float wave_incl_scan(float v, int lane) {
    #pragma unroll
    for (int d = 1; d < 32; d <<= 1) {
        float t = __shfl_up(v, d, 32);
        if (lane >= d) v += t;
    }
    return v;
}

// ---------------------------------------------------------------------------
// X-channel: blur (f32 WMMA) + row-wise diff/relu/prefix-sum + clips.
// Block = 16-row stripe x 512 cols of channel 0 of one batch. 512 thr = 16 waves.
// ---------------------------------------------------------------------------
#define PINP 520
__global__ __launch_bounds__(512) void grid_x_kernel(const float* __restrict__ pg,
                                                     float* __restrict__ gx,
                                                     float w0, float w1, float w2, float w3, float w4) {
    __shared__ float smem[20 * PINP];            // 41.6 KB; later overlaid by 16x512 stripe
    const int b    = blockIdx.y;
    const int y0   = blockIdx.x * 16;
    const int tid  = threadIdx.x;
    const int lane = tid & 31;
    const int wav  = tid >> 5;
    const int h    = lane >> 4;                  // half-wave select (K pairing)
    const int n    = lane & 15;                  // M/N index within tile

    const float* src = pg + (size_t)b * 2 * WW * WW;   // channel 0 plane

    // Stage 1: stage reflect-padded patch rows y0-2..y0+17, cols -2..513 into LDS
    for (int idx = tid; idx < 20 * 516; idx += 512) {
        int j = idx / 516, c = idx - j * 516;
        smem[j * PINP + c] = src[reflect_idx(y0 - 2 + j) * WW + reflect_idx(c - 2)];
    }
    __syncthreads();

    // Stage 2: per wave, two 16x16 tiles. Separable blur:
    //   D(16x16) = Lmat(16x20) x T(20x16), Lmat[m][j]=k1[j-m] (const A),
    //   T = horizontal 1D blur, computed on the fly into B fragments.
    v8f acc[2];
    #pragma unroll
    for (int t = 0; t < 2; ++t) {
        const int x0 = (wav * 2 + t) * 16;
        v8f c8 = {};
        #pragma unroll
        for (int k = 0; k < 5; ++k) {
            const int j0 = 4 * k + 2 * h;
            v2f a;
            a.x = wsel(j0     - n, w0, w1, w2, w3, w4);
            a.y = wsel(j0 + 1 - n, w0, w1, w2, w3, w4);
            const float* r0 = &smem[j0 * PINP + (x0 + n)];
            const float* r1 = r0 + PINP;
            v2f bf;
            bf.x = r0[0]*w0 + r0[1]*w1 + r0[2]*w2 + r0[3]*w3 + r0[4]*w4;
            bf.y = r1[0]*w0 + r1[1]*w1 + r1[2]*w2 + r1[3]*w3 + r1[4]*w4;
            c8 = __builtin_amdgcn_wmma_f32_16x16x4_f32(false, a, false, bf, (short)0, c8, false, false);
        }
        acc[t] = c8;
    }
    __syncthreads();

    // Stage 3: dump blurred tiles into stripe S[16][512] (overlaid on smem)
    float* S = smem;
    #pragma unroll
    for (int t = 0; t < 2; ++t) {
        const int x0 = (wav * 2 + t) * 16;
        #pragma unroll
        for (int v = 0; v < 8; ++v) {
            S[(v + 8 * h) * WW + (x0 + n)] = acc[t][v];
        }
    }
    __syncthreads();

    // Stage 4: wave 'wav' scans row y0+wav (512 elems = 16 chunks of 32)
    float carry = 0.0f, sx_last = 0.0f;
    const int y = y0 + wav;
    float* dst = gx + ((size_t)b * WW + y) * WW;
    for (int ch = 0; ch < 16; ++ch) {
        const int x  = ch * 32 + lane;
        float sx = S[wav * WW + x] + idenv(x) + kDEFOFF;
        float sp = __shfl_up(sx, 1, 32);
        if (lane == 0) sp = (ch == 0) ? 0.0f : sx_last;
        float a = fmaxf((sx - sp) / kSCALE, 0.0f) * kSCALE;   // relu(ax/SCALE)*SCALE
        float incl = wave_incl_scan(a, lane);
        float tot  = incl + carry;
        float dsp  = tot - kDEFOFF - idenv(x);
        dsp = fminf(fmaxf(dsp, -kCLIP), kCLIP);
        dst[x] = fminf(fmaxf(dsp + idenv(x), -1.0f), 1.0f);
        carry  += __shfl(incl, 31, 32);
        sx_last = __shfl(sx,   31, 32);
    }
}

// ---------------------------------------------------------------------------
// Y-channel: same, transposed. Block = 16-col stripe x 512 rows of channel 1.
// ---------------------------------------------------------------------------
#define PYP 20
#define SYP 17      // gcd(17,64)=1 -> conflict-free column scan
__global__ __launch_bounds__(512) void grid_y_kernel(const float* __restrict__ pg,
                                                     float* __restrict__ gy,
                                                     float w0, float w1, float w2, float w3, float w4) {
    __shared__ float smem[516 * PYP];            // 41.3 KB; overlaid by 512x17 stripe
    const int b    = blockIdx.y;
    const int X0   = blockIdx.x * 16;
    const int tid  = threadIdx.x;
    const int lane = tid & 31;
    const int wav  = tid >> 5;
    const int h    = lane >> 4;
    const int n    = lane & 15;

    const float* src = pg + ((size_t)b * 2 + 1) * WW * WW;   // channel 1 plane

    // Stage 1: patch rows -2..513, cols X0-2..X0+17 (reflect)
    for (int idx = tid; idx < 516 * 20; idx += 512) {
        int r = idx / 20, c = idx - r * 20;
        smem[r * PYP + c] = src[reflect_idx(r - 2) * WW + reflect_idx(X0 - 2 + c)];
    }
    __syncthreads();

    // Stage 2: WMMA tiles stacked vertically (2 per wave)
    v8f acc[2];
    #pragma unroll
    for (int t = 0; t < 2; ++t) {
        const int y0 = (wav * 2 + t) * 16;
        v8f c8 = {};
        #pragma unroll
        for (int k = 0; k < 5; ++k) {
            const int j0 = 4 * k + 2 * h;
            v2f a;
            a.x = wsel(j0     - n, w0, w1, w2, w3, w4);
            a.y = wsel(j0 + 1 - n, w0, w1, w2, w3, w4);
            const float* r0 = &smem[(y0 + j0) * PYP + n];
            const float* r1 = r0 + PYP;
            v2f bf;
            bf.x = r0[0]*w0 + r0[1]*w1 + r0[2]*w2 + r0[3]*w3 + r0[4]*w4;
            bf.y = r1[0]*w0 + r1[1]*w1 + r1[2]*w2 + r1[3]*w3 + r1[4]*w4;
            c8 = __builtin_amdgcn_wmma_f32_16x16x4_f32(false, a, false, bf, (short)0, c8, false, false);
        }
        acc[t] = c8;
    }
    __syncthreads();

    // Stage 3: stripe S[512][17] overlaid on smem
    float* S = smem;
    #pragma unroll
    for (int t = 0; t < 2; ++t) {
        const int yb = (wav * 2 + t) * 16;
        #pragma unroll
        for (int v = 0; v < 8; ++v) {
            S[(yb + v + 8 * h) * SYP + n] = acc[t][v];
        }
    }
    __syncthreads();

    // Stage 4: wave 'wav' scans column X0+wav down 512 rows; write back in place
    float carry = 0.0f, sy_last = 0.0f;
    for (int ch = 0; ch < 16; ++ch) {
        const int y = ch * 32 + lane;
        float sy = S[y * SYP + wav] + idenv(y) + kDEFOFF;
        float sp = __shfl_up(sy, 1, 32);
        if (lane == 0) sp = (ch == 0) ? 0.0f : sy_last;
        float a = fmaxf((sy - sp) / kSCALE, 0.0f) * kSCALE;
        float incl = wave_incl_scan(a, lane);
        float tot  = incl + carry;
        float dsp  = tot - kDEFOFF - idenv(y);
        dsp = fminf(fmaxf(dsp, -kCLIP), kCLIP);
        S[y * SYP + wav] = fminf(fmaxf(dsp + idenv(y), -1.0f), 1.0f);
        carry  += __shfl(incl, 31, 32);
        sy_last = __shfl(sy,   31, 32);
    }
    __syncthreads();

    // Stage 5: coalesced-ish writeback of the 512x16 stripe
    float* dst = gy + (size_t)b * WW * WW;
    for (int idx = tid; idx < WW * 16; idx += 512) {
        int y = idx >> 4, cx = idx & 15;
        dst[y * WW + X0 + cx] = S[y * SYP + cx];
    }
}

// ---------------------------------------------------------------------------
// Bilinear grid sample, 4 corners with validity masks (matches reference).
// ---------------------------------------------------------------------------
__global__ __launch_bounds__(256) void sample_kernel(const float* __restrict__ img,
                                                     const float* __restrict__ gx,
                                                     const float* __restrict__ gy,
                                                     float* __restrict__ out) {
    const size_t total = (size_t)NB * WW * WW;
    size_t idx = (size_t)blockIdx.x * blockDim.x + threadIdx.x;
    if (idx >= total) return;
    const int b  = (int)(idx >> 18);
    const int yx = (int)(idx & ((1u << 18) - 1));

    const float fx = (gx[idx] + 1.0f) * 0.5f * 511.0f;
    const float fy = (gy[idx] + 1.0f) * 0.5f * 511.0f;
    const float x0f = floorf(fx), y0f = floorf(fy);
    const float wx1 = fx - x0f, wx0 = 1.0f - wx1;
    const float wy1 = fy - y0f, wy0 = 1.0f - wy1;
    const int ix0 = (int)x0f, iy0 = (int)y0f;
    const int ix1 = ix0 + 1,  iy1 = iy0 + 1;

    const float mx0 = (ix0 >= 0 && ix0 < WW) ? 1.0f : 0.0f;
    const float mx1 = (ix1 >= 0 && ix1 < WW) ? 1.0f : 0.0f;
    const float my0 = (iy0 >= 0 && iy0 < WW) ? 1.0f : 0.0f;
    const float my1 = (iy1 >= 0 && iy1 < WW) ? 1.0f : 0.0f;
    const int cx0 = min(max(ix0, 0), WW - 1), cx1 = min(max(ix1, 0), WW - 1);
    const int cy0 = min(max(iy0, 0), WW - 1), cy1 = min(max(iy1, 0), WW - 1);

    const float w00 = wx0 * wy0 * mx0 * my0;
    const float w10 = wx1 * wy0 * mx1 * my0;
    const float w01 = wx0 * wy1 * mx0 * my1;
    const float w11 = wx1 * wy1 * mx1 * my1;

    const float* ib = img + (size_t)b * NC * WW * WW;
    #pragma unroll
    for (int c = 0; c < NC; ++c) {
        const float* pc = ib + (size_t)c * WW * WW;
        float v = pc[cy0 * WW + cx0] * w00 + pc[cy0 * WW + cx1] * w10
                + pc[cy1 * WW + cx0] * w01 + pc[cy1 * WW + cx1] * w11;
        out[((size_t)b * NC + c) * (WW * WW) + yx] = v;
    }
}

extern "C" void kernel_launch(void* const* d_in, const int* in_sizes, int n_in,
                              void* d_out, int out_size, void* d_ws, size_t ws_size,
                              hipStream_t stream) {
    const float* image = (const float*)d_in[0];
    const float* prim  = (const float*)d_in[1];
    float* out   = (float*)d_out;
    float* gridX = (float*)d_ws;                             // 32*512*512 f32
    float* gridY = gridX + (size_t)NB * WW * WW;             // 32*512*512 f32

    // Gaussian weights in double (matches numpy), cast to f32
    const double sigma = 5.0 * 0.15 + 0.35;
    double pdf[5], s = 0.0;
    for (int i = 0; i < 5; ++i) {
        double x = (double)(i - 2);
        pdf[i] = std::exp(-0.5 * (x / sigma) * (x / sigma));
        s += pdf[i];
    }
    const float w0 = (float)(pdf[0] / s), w1 = (float)(pdf[1] / s), w2 = (float)(pdf[2] / s),
                w3 = (float)(pdf[3] / s), w4 = (float)(pdf[4] / s);

    dim3 gblk(WW / 16, NB);
    grid_x_kernel<<<gblk, 512, 0, stream>>>(prim, gridX, w0, w1, w2, w3, w4);
    grid_y_kernel<<<gblk, 512, 0, stream>>>(prim, gridY, w0, w1, w2, w3, w4);

    const size_t total = (size_t)NB * WW * WW;
    sample_kernel<<<(unsigned)((total + 255) / 256), 256, 0, stream>>>(image, gridX, gridY, out);
}